// GraphSAGEModel_19000935317532
// MI455X (gfx1250) — compile-verified
//
#include <hip/hip_runtime.h>
#include <hip/hip_bf16.h>

// ---------------------------------------------------------------------------
// GraphSAGE / 3-layer GCN on MI455X (gfx1250, wave32).
// GEMM uses V_WMMA_F32_16X16X4_F32 (full fp32 precision; workload is
// bandwidth-bound so fp32 WMMA is free). W is staged in LDS in a K-pair-
// interleaved layout so each B fragment is one ds_load_b64 (no VGPR
// shuffles before the WMMAs). Edge aggregation uses global_atomic_add_f32
// (exactly jax.ops.segment_sum semantics).
// ---------------------------------------------------------------------------

#define NUM_USERS 100000
#define NUM_ITEMS 50000
#define NNODES    150000          // NUM_USERS + NUM_ITEMS
#define NEDGES    1200000
#define HID       64

typedef __attribute__((ext_vector_type(2))) float v2f;
typedef __attribute__((ext_vector_type(8))) float v8f;

// ---------------------------- small utility kernels ------------------------

__global__ void k_fill1(float* __restrict__ p, int n) {
  int i = blockIdx.x * blockDim.x + threadIdx.x;
  if (i < n) p[i] = 1.0f;                      // self-loop contribution to degree
}

__global__ void k_deg_accum(const int* __restrict__ dst, float* __restrict__ deg, int ne) {
  int e = blockIdx.x * blockDim.x + threadIdx.x;
  if (e < ne) atomicAdd(&deg[dst[e]], 1.0f);
}

__global__ void k_rsqrt_inplace(float* __restrict__ p, int n) {
  int i = blockIdx.x * blockDim.x + threadIdx.x;
  if (i < n) p[i] = rsqrtf(p[i]);              // deg >= 1 always (self loop)
}

__global__ void k_edge_norm(const int* __restrict__ src, const int* __restrict__ dst,
                            const float* __restrict__ dinv, float* __restrict__ nrm, int ne) {
  int e = blockIdx.x * blockDim.x + threadIdx.x;
  if (e < ne) nrm[e] = dinv[src[e]] * dinv[dst[e]];
}

__global__ void k_copy(const float* __restrict__ in, float* __restrict__ out, int n) {
  int i = blockIdx.x * blockDim.x + threadIdx.x;
  if (i < n) out[i] = in[i];
}

// ------------------------------- WMMA GEMM ---------------------------------
// H[r, :] = X[r, :] @ W  for 64x64 W, fp32, one 16-row tile per wave.
// Input rows come from a virtual concat: r < nu -> XU[r], else XI[r-nu].
//
// LDS layout for W: K-pair interleaved. Wp[kp*64 + n] = {W[2kp][n], W[2kp+1][n]}
// so the B fragment for (k, hi, n) is the single 8-byte word Wp[(k/2+hi)*64+n],
// loaded with one ds_load_b64 straight into an even-aligned VGPR pair.
__global__ void k_gemm_wmma(const float* __restrict__ XU, const float* __restrict__ XI,
                            const float* __restrict__ W, float* __restrict__ Hout,
                            int nu, int nrows) {
  __shared__ v2f Wp[(HID / 2) * HID];          // 32 k-pairs x 64 cols = 16 KB
  for (int i = threadIdx.x; i < HID * HID; i += blockDim.x) {
    int k = i >> 6;                            // source row (K)
    int n = i & 63;                            // source col (N)
    // scalar element write into the interleaved pair layout
    ((float*)Wp)[((k >> 1) * HID + n) * 2 + (k & 1)] = W[i];
  }
  __syncthreads();

  const int lane = threadIdx.x & 31;
  const int tile = blockIdx.x * (blockDim.x >> 5) + (threadIdx.x >> 5);
  const int r0 = tile * 16;
  if (r0 >= nrows) return;

  const int m  = lane & 15;    // M index (A rows / B,C,D column)
  const int hi = lane >> 4;    // half-wave selector (K split / M split for C,D)

  const int r = r0 + m;
  const float* xrow = (r < nu) ? (XU + (size_t)r * HID)
                               : (XI + (size_t)(r - nu) * HID);

  v8f acc0 = {}, acc1 = {}, acc2 = {}, acc3 = {};

#pragma unroll
  for (int k = 0; k < HID; k += 4) {
    // A fragment (16x4 f32): lanes 0-15 hold K={k,k+1}, lanes 16-31 K={k+2,k+3}
    v2f a = *(const v2f*)(xrow + k + 2 * hi);
    // B fragments (4x16 f32): same K split; one ds_load_b64 per column tile.
    const v2f* wrow = Wp + ((k >> 1) + hi) * HID;
    v2f b0 = wrow[ 0 + m];
    v2f b1 = wrow[16 + m];
    v2f b2 = wrow[32 + m];
    v2f b3 = wrow[48 + m];

    acc0 = __builtin_amdgcn_wmma_f32_16x16x4_f32(false, a, false, b0, (short)0, acc0, false, false);
    acc1 = __builtin_amdgcn_wmma_f32_16x16x4_f32(false, a, false, b1, (short)0, acc1, false, false);
    acc2 = __builtin_amdgcn_wmma_f32_16x16x4_f32(false, a, false, b2, (short)0, acc2, false, false);
    acc3 = __builtin_amdgcn_wmma_f32_16x16x4_f32(false, a, false, b3, (short)0, acc3, false, false);
  }

  // C/D layout: VGPR v -> row r0 + v + 8*hi, column = tile*16 + m
#pragma unroll
  for (int v = 0; v < 8; v++) {
    float* hr = Hout + (size_t)(r0 + v + 8 * hi) * HID;
    hr[ 0 + m] = acc0[v];
    hr[16 + m] = acc1[v];
    hr[32 + m] = acc2[v];
    hr[48 + m] = acc3[v];
  }
}

// --------------------- bias + self-loop init (full overwrite) --------------
// out[i,f] = bias[f] + H[i,f] * dinv[i]^2   (self-loop edge has norm dinv^2)
__global__ void k_init_out(const float* __restrict__ H, const float* __restrict__ dinv,
                           const float* __restrict__ bias,
                           float* __restrict__ outU, float* __restrict__ outI,
                           int nu, int n) {
  int gid = blockIdx.x * blockDim.x + threadIdx.x;
  if (gid >= n * HID) return;
  int node = gid >> 6;
  int f    = gid & 63;
  float di = dinv[node];
  float v  = bias[f] + H[gid] * di * di;
  float* orow = (node < nu) ? (outU + (size_t)node * HID)
                            : (outI + (size_t)(node - nu) * HID);
  orow[f] = v;
}

// ----------------------------- edge scatter --------------------------------
// One wave per edge: out[dst, :] += H[src, :] * norm[e]  (64 atomics, two
// coalesced 128B segments per row).
__global__ void k_scatter(const int* __restrict__ src, const int* __restrict__ dst,
                          const float* __restrict__ nrm, const float* __restrict__ H,
                          float* __restrict__ outU, float* __restrict__ outI,
                          int nu, int ne) {
  int e    = (blockIdx.x * blockDim.x + threadIdx.x) >> 5;
  int lane = threadIdx.x & 31;
  if (e >= ne) return;
  int s = src[e];
  int d = dst[e];
  float w = nrm[e];
  const float* hrow = H + (size_t)s * HID;
  float* orow = (d < nu) ? (outU + (size_t)d * HID)
                         : (outI + (size_t)(d - nu) * HID);
  atomicAdd(orow + lane,      hrow[lane]      * w);
  atomicAdd(orow + lane + 32, hrow[lane + 32] * w);
}

// ------------------------------- launcher ----------------------------------

extern "C" void kernel_launch(void* const* d_in, const int* in_sizes, int n_in,
                              void* d_out, int out_size, void* d_ws, size_t ws_size,
                              hipStream_t stream) {
  const int*   edge  = (const int*)d_in[0];          // (2, NEDGES) flat
  const float* uemb  = (const float*)d_in[1];        // (NUM_USERS, 64)
  const float* iemb  = (const float*)d_in[2];        // (NUM_ITEMS, 64)
  const float* Wsall = (const float*)d_in[3];        // (3, 64, 64)
  const float* bsall = (const float*)d_in[4];        // (3, 64)
  float*       out   = (float*)d_out;

  const int* src = edge;
  const int* dst = edge + NEDGES;

  // workspace layout (floats): dinv | norm | H | X   (~82 MB total)
  float* ws   = (float*)d_ws;
  float* dinv = ws;                                   // NNODES (deg -> dinv in place)
  float* nrm  = dinv + 150016;                        // NEDGES
  float* Hbuf = nrm + NEDGES;                         // NNODES*64
  float* Xbuf = Hbuf + (size_t)NNODES * HID;          // NNODES*64

  // d_out slots: [x_users | users_emb | x_items | items_emb]
  float* out_xu = out;
  float* out_ue = out + (size_t)NUM_USERS * HID;
  float* out_xi = out + (size_t)2 * NUM_USERS * HID;
  float* out_ie = out + (size_t)2 * NUM_USERS * HID + (size_t)NUM_ITEMS * HID;

  const int T = 256;
  const int gN   = (NNODES + T - 1) / T;
  const int gE   = (NEDGES + T - 1) / T;
  const int gNH  = (NNODES * HID + T - 1) / T;
  const int gSc  = (NEDGES * 32 + T - 1) / T;          // one wave per edge
  const int gGemm = ((NNODES / 16) + (T / 32) - 1) / (T / 32);

  // ---- graph-constant normalization ----
  k_fill1        <<<gN, T, 0, stream>>>(dinv, NNODES);
  k_deg_accum    <<<gE, T, 0, stream>>>(dst, dinv, NEDGES);
  k_rsqrt_inplace<<<gN, T, 0, stream>>>(dinv, NNODES);
  k_edge_norm    <<<gE, T, 0, stream>>>(src, dst, dinv, nrm, NEDGES);

  // ---- embedding pass-through outputs (independent regions) ----
  k_copy<<<(NUM_USERS * HID + T - 1) / T, T, 0, stream>>>(uemb, out_ue, NUM_USERS * HID);
  k_copy<<<(NUM_ITEMS * HID + T - 1) / T, T, 0, stream>>>(iemb, out_ie, NUM_ITEMS * HID);

  // ---- layer 0: inputs are the virtual concat [users_emb; items_emb] ----
  k_gemm_wmma<<<gGemm, T, 0, stream>>>(uemb, iemb, Wsall + 0 * HID * HID, Hbuf, NUM_USERS, NNODES);
  k_init_out <<<gNH, T, 0, stream>>>(Hbuf, dinv, bsall + 0 * HID, Xbuf, Xbuf + (size_t)NUM_USERS * HID, NUM_USERS, NNODES);
  k_scatter  <<<gSc, T, 0, stream>>>(src, dst, nrm, Hbuf, Xbuf, Xbuf + (size_t)NUM_USERS * HID, NUM_USERS, NEDGES);

  // ---- layer 1: X -> H -> X (X is dead after its GEMM, safe to reuse) ----
  k_gemm_wmma<<<gGemm, T, 0, stream>>>(Xbuf, Xbuf, Wsall + 1 * HID * HID, Hbuf, NNODES, NNODES);
  k_init_out <<<gNH, T, 0, stream>>>(Hbuf, dinv, bsall + 1 * HID, Xbuf, Xbuf + (size_t)NUM_USERS * HID, NUM_USERS, NNODES);
  k_scatter  <<<gSc, T, 0, stream>>>(src, dst, nrm, Hbuf, Xbuf, Xbuf + (size_t)NUM_USERS * HID, NUM_USERS, NEDGES);

  // ---- layer 2: X -> H -> d_out (split users/items destinations) ----
  k_gemm_wmma<<<gGemm, T, 0, stream>>>(Xbuf, Xbuf, Wsall + 2 * HID * HID, Hbuf, NNODES, NNODES);
  k_init_out <<<gNH, T, 0, stream>>>(Hbuf, dinv, bsall + 2 * HID, out_xu, out_xi, NUM_USERS, NNODES);
  k_scatter  <<<gSc, T, 0, stream>>>(src, dst, nrm, Hbuf, out_xu, out_xi, NUM_USERS, NEDGES);
}